// SimpleTGNModel_16372415332401
// MI455X (gfx1250) — compile-verified
//
#include <hip/hip_runtime.h>

typedef __attribute__((ext_vector_type(2))) float v2f;
typedef __attribute__((ext_vector_type(8))) float v8f;

#define H     128
#define EDGE  64
#define TD    16
#define GIN   336     // 2H + EDGE + TD
#define LDI   344     // padded LDS row stride (floats) for input tile, 8-float aligned
#define LDH   264     // padded stride for predictor h tile (2H=256)
#define LDR   132     // padded stride for predictor relu tile (H=128)

__device__ __forceinline__ float fast_sigmoid(float x) {
    return 1.0f / (1.0f + __expf(-x));
}
__device__ __forceinline__ float fast_tanh(float x) {
    float ax = fabsf(x);
    float e  = __expf(-2.0f * ax);
    float t  = (1.0f - e) / (1.0f + e);
    return copysignf(t, x);
}

// One 16x16 f32 tile via V_WMMA_F32_16X16X4_F32 over K (multiple of 4).
// ap: this lane's A row base  = &A[lane&15][khi]        (row stride pre-applied)
// wp: this lane's B col base  = &W[(n0 + (lane&15))*ldw + khi]   (B[k][n] = W[n][k])
__device__ __forceinline__ v8f wmma_k_loop(const float* ap, const float* __restrict__ wp, int K) {
    v8f acc = {};
    for (int k = 0; k < K; k += 4) {
        v2f a = *(const v2f*)(ap + k);
        v2f b = *(const v2f*)(wp + k);
        acc = __builtin_amdgcn_wmma_f32_16x16x4_f32(false, a, false, b, (short)0, acc,
                                                    false, false);
    }
    return acc;
}

// ---------------------------------------------------------------- K1: copy
__global__ __launch_bounds__(256) void tgn_copy_kernel(float* __restrict__ outp,
                                                       const float* __restrict__ inp,
                                                       long n4) {
    long i = (long)blockIdx.x * blockDim.x + threadIdx.x;
    if (i < n4) ((float4*)outp)[i] = ((const float4*)inp)[i];
}

// ---------------------------------------------------------------- K2: GRU per event tile
__global__ __launch_bounds__(256) void tgn_gru_kernel(
    const float* __restrict__ memory, const int* __restrict__ src, const int* __restrict__ dst,
    const float* __restrict__ ts, const float* __restrict__ ef,
    const float* __restrict__ Wt, const float* __restrict__ bt,
    const float* __restrict__ Wih, const float* __restrict__ Whh,
    const float* __restrict__ bih, const float* __restrict__ bhh,
    float* __restrict__ ws_s, float* __restrict__ ws_d, int E)
{
    __shared__ float sInp[16 * LDI];   // [16 rows][336 cols + pad] : s | d | ef | te
    const int t  = threadIdx.x;
    const int e0 = blockIdx.x * 16;

    // ---- Phase A: cooperatively build the [16 x 336] input tile ----
    for (int idx = t; idx < 16 * 32; idx += 256) {       // s and d: 16 rows x 32 float4
        int row = idx >> 5, c4 = (idx & 31) << 2;
        int ev  = min(e0 + row, E - 1);
        float4 sv = *(const float4*)(memory + (size_t)src[ev] * H + c4);
        float4 dv = *(const float4*)(memory + (size_t)dst[ev] * H + c4);
        *(float4*)(sInp + row * LDI + c4)     = sv;
        *(float4*)(sInp + row * LDI + H + c4) = dv;
    }
    for (int idx = t; idx < 16 * 16; idx += 256) {       // ef: 16 rows x 16 float4
        int row = idx >> 4, c4 = (idx & 15) << 2;
        int ev  = min(e0 + row, E - 1);
        *(float4*)(sInp + row * LDI + 2 * H + c4) =
            *(const float4*)(ef + (size_t)ev * EDGE + c4);
    }
    if (t < 16 * TD) {                                   // time encoding
        int row = t >> 4, col = t & 15;
        int ev  = min(e0 + row, E - 1);
        sInp[row * LDI + 2 * H + EDGE + col] = sinf(ts[ev] * Wt[col] + bt[col]);
    }
    __syncthreads();

    // ---- Phase B: each wave owns output cols [16w, 16w+16) ----
    const int lane = t & 31;
    const int w    = t >> 5;
    const int r    = lane & 15;
    const int khi  = (lane >> 4) << 1;        // 0 or 2 (A/B f32 16x16x4 layout)
    const int n0   = w * 16;
    const int ncol = n0 + r;                  // this lane's N column (0..127)

    const float* aFull = sInp + r * LDI + khi;        // inp rows (K over 336)
    const float* aS    = aFull;                       // s = cols [0,128)
    const float* aD    = sInp + r * LDI + H + khi;    // d = cols [128,256)

    // r gate
    v8f i_r  = wmma_k_loop(aFull, Wih + (size_t)(0 * H + ncol) * GIN + khi, GIN);
    v8f hr_s = wmma_k_loop(aS,    Whh + (size_t)(0 * H + ncol) * H   + khi, H);
    v8f hr_d = wmma_k_loop(aD,    Whh + (size_t)(0 * H + ncol) * H   + khi, H);
    float b_r = bih[ncol] + bhh[ncol];
    v8f r_s, r_d;
    #pragma unroll
    for (int i = 0; i < 8; ++i) {
        r_s[i] = fast_sigmoid(i_r[i] + hr_s[i] + b_r);
        r_d[i] = fast_sigmoid(i_r[i] + hr_d[i] + b_r);
    }

    // z gate
    v8f i_z  = wmma_k_loop(aFull, Wih + (size_t)(1 * H + ncol) * GIN + khi, GIN);
    v8f hz_s = wmma_k_loop(aS,    Whh + (size_t)(1 * H + ncol) * H   + khi, H);
    v8f hz_d = wmma_k_loop(aD,    Whh + (size_t)(1 * H + ncol) * H   + khi, H);
    float b_z = bih[H + ncol] + bhh[H + ncol];
    v8f z_s, z_d;
    #pragma unroll
    for (int i = 0; i < 8; ++i) {
        z_s[i] = fast_sigmoid(i_z[i] + hz_s[i] + b_z);
        z_d[i] = fast_sigmoid(i_z[i] + hz_d[i] + b_z);
    }

    // n gate + output
    v8f i_n  = wmma_k_loop(aFull, Wih + (size_t)(2 * H + ncol) * GIN + khi, GIN);
    v8f hn_s = wmma_k_loop(aS,    Whh + (size_t)(2 * H + ncol) * H   + khi, H);
    v8f hn_d = wmma_k_loop(aD,    Whh + (size_t)(2 * H + ncol) * H   + khi, H);
    float b_in = bih[2 * H + ncol];
    float b_hn = bhh[2 * H + ncol];

    const int Mbase = (lane >> 4) << 3;       // C/D layout: M = vgpr + 8*(lane>=16)
    #pragma unroll
    for (int i = 0; i < 8; ++i) {
        int M  = Mbase + i;
        int ev = e0 + M;
        float hs = sInp[M * LDI + n0 + r];        // old s state, this column
        float hd = sInp[M * LDI + H + n0 + r];    // old d state
        float ns = fast_tanh(i_n[i] + b_in + r_s[i] * (hn_s[i] + b_hn));
        float nd = fast_tanh(i_n[i] + b_in + r_d[i] * (hn_d[i] + b_hn));
        float os = (1.0f - z_s[i]) * ns + z_s[i] * hs;
        float od = (1.0f - z_d[i]) * nd + z_d[i] * hd;
        if (ev < E) {
            ws_s[(size_t)ev * H + n0 + r] = os;
            ws_d[(size_t)ev * H + n0 + r] = od;
        }
    }
}

// ---------------------------------------------------------------- K3/K4: scatter
__global__ __launch_bounds__(256) void tgn_scatter_kernel(float* __restrict__ mem_out,
                                                          const float* __restrict__ vals,
                                                          const int* __restrict__ idx, int E) {
    int i = blockIdx.x * blockDim.x + threadIdx.x;     // over E * 32 float4 chunks
    if (i >= E * (H / 4)) return;
    int ev = i >> 5;
    int c4 = (i & 31) << 2;
    float4 v = *(const float4*)(vals + (size_t)ev * H + c4);
    *(float4*)(mem_out + (size_t)idx[ev] * H + c4) = v;
}

// ---------------------------------------------------------------- K5: predictor
__global__ __launch_bounds__(256) void tgn_pred_kernel(
    const float* __restrict__ mem, const int* __restrict__ src, const int* __restrict__ dst,
    const float* __restrict__ Wp1, const float* __restrict__ bp1,
    const float* __restrict__ Wp2, const float* __restrict__ bp2,
    float* __restrict__ pred, int E)
{
    __shared__ float sH[16 * LDH];    // [16][256+pad] : updated src | dst states
    __shared__ float sR[16 * LDR];    // [16][128+pad] : relu(layer1)
    const int t  = threadIdx.x;
    const int e0 = blockIdx.x * 16;

    for (int idx = t; idx < 16 * 32; idx += 256) {
        int row = idx >> 5, c4 = (idx & 31) << 2;
        int ev  = min(e0 + row, E - 1);
        float4 sv = *(const float4*)(mem + (size_t)src[ev] * H + c4);
        float4 dv = *(const float4*)(mem + (size_t)dst[ev] * H + c4);
        *(float4*)(sH + row * LDH + c4)     = sv;
        *(float4*)(sH + row * LDH + H + c4) = dv;
    }
    __syncthreads();

    const int lane = t & 31;
    const int w    = t >> 5;
    const int r    = lane & 15;
    const int khi  = (lane >> 4) << 1;
    const int n0   = w * 16;

    v8f acc = wmma_k_loop(sH + r * LDH + khi,
                          Wp1 + (size_t)(n0 + r) * (2 * H) + khi, 2 * H);
    float b1 = bp1[n0 + r];
    const int Mbase = (lane >> 4) << 3;
    #pragma unroll
    for (int i = 0; i < 8; ++i) {
        int M = Mbase + i;
        sR[M * LDR + n0 + r] = fmaxf(acc[i] + b1, 0.0f);
    }
    __syncthreads();

    if (t < 16) {
        int ev = e0 + t;
        if (ev < E) {
            float sum = bp2[0];
            for (int c = 0; c < H; ++c) sum += sR[t * LDR + c] * Wp2[c];
            pred[ev] = sum;
        }
    }
}

// ---------------------------------------------------------------- host launch
extern "C" void kernel_launch(void* const* d_in, const int* in_sizes, int n_in,
                              void* d_out, int out_size, void* d_ws, size_t ws_size,
                              hipStream_t stream) {
    const float* memory = (const float*)d_in[0];
    const int*   src    = (const int*)d_in[1];
    const int*   dst    = (const int*)d_in[2];
    const float* ts     = (const float*)d_in[3];
    const float* ef     = (const float*)d_in[4];
    const float* Wt     = (const float*)d_in[5];
    const float* bt     = (const float*)d_in[6];
    const float* Wih    = (const float*)d_in[7];
    const float* Whh    = (const float*)d_in[8];
    const float* bih    = (const float*)d_in[9];
    const float* bhh    = (const float*)d_in[10];
    const float* Wp1    = (const float*)d_in[11];
    const float* bp1    = (const float*)d_in[12];
    const float* Wp2    = (const float*)d_in[13];
    const float* bp2    = (const float*)d_in[14];

    const int  E        = in_sizes[1];
    const long memElems = in_sizes[0];          // N_NODES * 128

    float* pred   = (float*)d_out;
    float* memOut = (float*)d_out + E;
    float* ws_s   = (float*)d_ws;
    float* ws_d   = ws_s + (size_t)E * H;

    // K1: copy original memory into output region
    long n4 = memElems / 4;
    unsigned cg = (unsigned)((n4 + 255) / 256);
    tgn_copy_kernel<<<cg, 256, 0, stream>>>(memOut, memory, n4);

    // K2: per-event GRU updates (reads ORIGINAL memory) -> workspace
    int nTiles = (E + 15) / 16;
    tgn_gru_kernel<<<nTiles, 256, 0, stream>>>(memory, src, dst, ts, ef, Wt, bt,
                                               Wih, Whh, bih, bhh, ws_s, ws_d, E);

    // K3/K4: scatter src updates, then dst updates (dst wins, matching reference)
    int sThreads = E * (H / 4);
    int sGrid    = (sThreads + 255) / 256;
    tgn_scatter_kernel<<<sGrid, 256, 0, stream>>>(memOut, ws_s, src, E);
    tgn_scatter_kernel<<<sGrid, 256, 0, stream>>>(memOut, ws_d, dst, E);

    // K5: predictor on updated memory
    tgn_pred_kernel<<<nTiles, 256, 0, stream>>>(memOut, src, dst, Wp1, bp1, Wp2, bp2,
                                                pred, E);
}